// GridSmoother_41360535060648
// MI455X (gfx1250) — compile-verified
//
#include <hip/hip_runtime.h>

// Problem constants (from reference setup_inputs)
#define B_   16
#define D_   16
#define H_   128
#define W_   160
#define NPLANE (H_ * W_)
#define NTOT ((size_t)B_ * D_ * NPLANE)   // 5,242,880 floats = 21 MB / vector
#define MAXITER 500

// Spatial tile per block: 16x32 points, 256 threads (8 wave32s), 2 points per
// thread (rows lh and lh+8), looping over the 16 D-planes (weights reused 16x).
#define TH 16
#define TW 32
#define LW_ (TW + 2)                      // 34
#define LSZ ((TH + 2) * LW_)              // 612 floats of halo'd tile in LDS

// k_update / k_xfinal: 640 blocks x 256 threads x 8 float4 = NTOT/4 exactly
#define UPD_BLOCKS 640
#define UPD_CHUNK  8
#define UPD_STRIDE ((size_t)UPD_BLOCKS * 256)

// double accumulator slots in d_ws
#define ACC_ZERO   0                       // stays 0.0 (alpha/beta = 0 at k=0)
#define ACC_RR(k)  (1 + (k))               // rr_k, k = 0..MAXITER
#define ACC_PAP(k) (600 + (k))             // p.Ap at iter k, k = 0..MAXITER-1
#define ACC_COUNT  1104

typedef float v2f __attribute__((ext_vector_type(2)));
typedef float v8f __attribute__((ext_vector_type(8)));

// ---------------------------------------------------------------------------
// Wave32 all-reduce via the matrix core: two V_WMMA_F32_16X16X4_F32.
// A holds one partial per lane (VGPR0) and 0 (VGPR1); B = ones.
// WMMA#1: D[m][n] = rowsum(m) -> each lane sums its 8 D VGPRs (half-total).
// WMMA#2 with the half-totals as A broadcasts the full 32-lane sum to all lanes.
// Full fp32 precision; requires EXEC all-ones (call from uniform code).
// ---------------------------------------------------------------------------
__device__ __forceinline__ float wave_allreduce_wmma(float v) {
  v2f ones; ones.x = 1.0f; ones.y = 1.0f;
  v2f a;    a.x = v;       a.y = 0.0f;
  v8f c = {};
  c = __builtin_amdgcn_wmma_f32_16x16x4_f32(false, a, false, ones, (short)0, c,
                                            false, false);
  float t = c[0] + c[1] + c[2] + c[3] + c[4] + c[5] + c[6] + c[7];
  v2f a2; a2.x = t; a2.y = 0.0f;
  v8f c2 = {};
  c2 = __builtin_amdgcn_wmma_f32_16x16x4_f32(false, a2, false, ones, (short)0, c2,
                                             false, false);
  return c2[0];
}

// Block-level accumulation of a per-thread partial into a global double slot.
__device__ __forceinline__ void block_accum(float v, double* slot, double* red) {
  float w = wave_allreduce_wmma(v);   // every lane holds the wave sum
  int tid = threadIdx.x;
  if ((tid & 31) == 0) red[tid >> 5] = (double)w;
  __syncthreads();
  if (tid == 0) {
    double tot = 0.0;
    for (int i = 0; i < 8; ++i) tot += red[i];   // 256 threads = 8 waves
    atomicAdd(slot, tot);
  }
}

// ---------------------------------------------------------------------------
__global__ __launch_bounds__(256) void k_zero(double* acc) {
  for (int i = threadIdx.x; i < ACC_COUNT; i += 256) acc[i] = 0.0;
}

// Per-thread boundary-masked edge weights (zero weight kills out-of-range terms,
// so clamped halo loads only need to be *finite* at the plane boundary).
__device__ __forceinline__ void load_weights(const float* wx, const float* wy,
                                             int h, int w, float& wL, float& wR,
                                             float& wU, float& wDn) {
  wL  = (w > 0)      ? wx[h * W_ + (w - 1)] : 0.0f;
  wR  = (w < W_ - 1) ? wx[h * W_ + w]       : 0.0f;
  wU  = (h > 0)      ? wy[(h - 1) * W_ + w] : 0.0f;
  wDn = (h < H_ - 1) ? wy[h * W_ + w]       : 0.0f;
}

__device__ __forceinline__ int clampi(int v, int lo, int hi) {
  return v < lo ? lo : (v > hi ? hi : v);
}

// lap from the halo'd LDS tile at center index c.
__device__ __forceinline__ float stencil_lap(const float* s, int c, float wL,
                                             float wR, float wU, float wDn) {
  float ctr = s[c];
  return wL * (ctr - s[c - 1]) - wR * (s[c + 1] - ctr) +
         wU * (ctr - s[c - LW_]) - wDn * (s[c + LW_] - ctr);
}

// Branchless clamped global index for halo element t of the tile.
__device__ __forceinline__ size_t halo_gi(int t, int th0, int tw0, size_t plane) {
  int hh = t / LW_, ww = t % LW_;
  int gh = clampi(th0 - 1 + hh, 0, H_ - 1);
  int gw = clampi(tw0 - 1 + ww, 0, W_ - 1);
  return plane + (size_t)gh * W_ + gw;
}

// ---------------------------------------------------------------------------
// init: x = b (= ae), r = b - A(b) = -lap(b), p0 = r, acc[RR(0)] += r.r
// ---------------------------------------------------------------------------
__global__ __launch_bounds__(256) void k_init(const float* __restrict__ ae,
                                              const float* __restrict__ wxwy,
                                              float* __restrict__ x,
                                              float* __restrict__ r,
                                              float* __restrict__ p0,
                                              double* __restrict__ acc) {
  __shared__ float s[LSZ];
  __shared__ double red[8];
  int tid = threadIdx.x;
  int tw0 = blockIdx.x * TW, th0 = blockIdx.y * TH, b = blockIdx.z;
  int lw = tid & 31, lh = tid >> 5;               // 8 sub-rows of 32 lanes
  int w = tw0 + lw;
  const float* wx = wxwy + (size_t)(b * 2 + 0) * NPLANE;
  const float* wy = wxwy + (size_t)(b * 2 + 1) * NPLANE;
  float wL[2], wR[2], wU[2], wDn[2];
  int hpt[2], cpt[2];
  for (int j = 0; j < 2; ++j) {                   // 2 points/thread: lh, lh+8
    hpt[j] = th0 + lh + j * 8;
    cpt[j] = (lh + 1 + j * 8) * LW_ + (lw + 1);
    load_weights(wx, wy, hpt[j], w, wL[j], wR[j], wU[j], wDn[j]);
  }
  float rr = 0.0f;

  for (int d = 0; d < D_; ++d) {
    size_t plane = (size_t)(b * D_ + d) * NPLANE;
    // unrolled staging: 612 = 256 + 256 + 100 (loads issued before use)
    s[tid]       = ae[halo_gi(tid,       th0, tw0, plane)];
    s[tid + 256] = ae[halo_gi(tid + 256, th0, tw0, plane)];
    if (tid < LSZ - 512) s[tid + 512] = ae[halo_gi(tid + 512, th0, tw0, plane)];
    __syncthreads();
    for (int j = 0; j < 2; ++j) {
      float rv = -stencil_lap(s, cpt[j], wL[j], wR[j], wU[j], wDn[j]);
      size_t idx = plane + (size_t)hpt[j] * W_ + w;
      x[idx]  = s[cpt[j]];
      r[idx]  = rv;
      p0[idx] = rv;
      rr += rv * rv;
    }
    __syncthreads();
  }
  block_accum(rr, &acc[ACC_RR(0)], red);
}

// ---------------------------------------------------------------------------
// Fused kernel at iteration k (ping-pong p to stay race-free):
//   beta       = acc[bn]/acc[bd]        (0 at k=0 via ACC_ZERO slot)
//   alpha_prev = acc[an]/acc[ad]        (0 at k=0)
//   p_new = r + beta*p_in  (staged in LDS with halo; p_in/r are READ-ONLY here)
//   write p_out = p_new ; Ap = p_new + lap(p_new) ; acc[pap] += p_new . Ap
//   lagged solution update: x += alpha_prev * p_in   (k>0)
// ---------------------------------------------------------------------------
__global__ __launch_bounds__(256) void k_matvec(const float* __restrict__ r,
                                                const float* __restrict__ p_in,
                                                float* __restrict__ p_out,
                                                float* __restrict__ Ap,
                                                float* __restrict__ x,
                                                const float* __restrict__ wxwy,
                                                double* __restrict__ acc,
                                                int bn, int bd, int an, int ad,
                                                int pap_idx, int do_x) {
  __shared__ float s[LSZ];
  __shared__ double red[8];
  int tid = threadIdx.x;
  int tw0 = blockIdx.x * TW, th0 = blockIdx.y * TH, b = blockIdx.z;
  int lw = tid & 31, lh = tid >> 5;
  int w = tw0 + lw;
  const float* wx = wxwy + (size_t)(b * 2 + 0) * NPLANE;
  const float* wy = wxwy + (size_t)(b * 2 + 1) * NPLANE;
  float wL[2], wR[2], wU[2], wDn[2];
  int hpt[2], cpt[2];
  for (int j = 0; j < 2; ++j) {
    hpt[j] = th0 + lh + j * 8;
    cpt[j] = (lh + 1 + j * 8) * LW_ + (lw + 1);
    load_weights(wx, wy, hpt[j], w, wL[j], wR[j], wU[j], wDn[j]);
  }
  float beta       = (float)(acc[bn] / acc[bd]);
  float alpha_prev = (float)(acc[an] / acc[ad]);
  float pap = 0.0f;

  for (int d = 0; d < D_; ++d) {
    size_t plane = (size_t)(b * D_ + d) * NPLANE;
    {
      size_t g0 = halo_gi(tid,       th0, tw0, plane);
      size_t g1 = halo_gi(tid + 256, th0, tw0, plane);
      float r0 = r[g0], p0v = p_in[g0];
      float r1 = r[g1], p1v = p_in[g1];
      s[tid]       = r0 + beta * p0v;
      s[tid + 256] = r1 + beta * p1v;
      if (tid < LSZ - 512) {
        size_t g2 = halo_gi(tid + 512, th0, tw0, plane);
        s[tid + 512] = r[g2] + beta * p_in[g2];
      }
    }
    __syncthreads();
    for (int j = 0; j < 2; ++j) {
      float pn = s[cpt[j]];
      float ap = pn + stencil_lap(s, cpt[j], wL[j], wR[j], wU[j], wDn[j]);
      size_t idx = plane + (size_t)hpt[j] * W_ + w;
      p_out[idx] = pn;
      Ap[idx]    = ap;
      pap += pn * ap;
      if (do_x)                                // uniform branch, EXEC stays full
        x[idx] += alpha_prev * p_in[idx];      // lagged x += a_{k-1} p_{k-1}
    }
    __syncthreads();
  }
  block_accum(pap, &acc[pap_idx], red);
}

// ---------------------------------------------------------------------------
// Residual update for iteration k (pointwise, race-free):
//   alpha = rr_k / pAp_k ; r -= alpha Ap ; acc[rr_{k+1}] += r.r
// 640 blocks x 256 threads x 8 float4 = exact cover; fewer same-address atomics.
// ---------------------------------------------------------------------------
__global__ __launch_bounds__(256) void k_update(float* __restrict__ r,
                                                const float* __restrict__ Ap,
                                                double* __restrict__ acc,
                                                int rrk_idx, int pap_idx,
                                                int rrn_idx) {
  __shared__ double red[8];
  float alpha = (float)(acc[rrk_idx] / acc[pap_idx]);
  float rr = 0.0f;
  size_t gi = (size_t)blockIdx.x * 256 + threadIdx.x;   // float4 index
  for (int j = 0; j < UPD_CHUNK; ++j, gi += UPD_STRIDE) {
    float4 av = reinterpret_cast<const float4*>(Ap)[gi];
    float4 rv = reinterpret_cast<float4*>(r)[gi];
    rv.x -= alpha * av.x; rv.y -= alpha * av.y;
    rv.z -= alpha * av.z; rv.w -= alpha * av.w;
    reinterpret_cast<float4*>(r)[gi] = rv;
    rr += rv.x * rv.x + rv.y * rv.y + rv.z * rv.z + rv.w * rv.w;
  }
  block_accum(rr, &acc[rrn_idx], red);
}

// Final lagged term: x += alpha_{last} * p_{last}
__global__ __launch_bounds__(256) void k_xfinal(float* __restrict__ x,
                                                const float* __restrict__ p,
                                                double* __restrict__ acc,
                                                int rrk_idx, int pap_idx) {
  float alpha = (float)(acc[rrk_idx] / acc[pap_idx]);
  size_t gi = (size_t)blockIdx.x * 256 + threadIdx.x;   // float4 index
  for (int j = 0; j < UPD_CHUNK; ++j, gi += UPD_STRIDE) {
    float4 pv = reinterpret_cast<const float4*>(p)[gi];
    float4 xv = reinterpret_cast<float4*>(x)[gi];
    xv.x += alpha * pv.x; xv.y += alpha * pv.y;
    xv.z += alpha * pv.z; xv.w += alpha * pv.w;
    reinterpret_cast<float4*>(x)[gi] = xv;
  }
}

// ---------------------------------------------------------------------------
extern "C" void kernel_launch(void* const* d_in, const int* in_sizes, int n_in,
                              void* d_out, int out_size, void* d_ws,
                              size_t ws_size, hipStream_t stream) {
  const float* ae   = (const float*)d_in[0];   // (B,D,H,W) fp32
  const float* wxwy = (const float*)d_in[1];   // (B,2,H,W) fp32
  float* x = (float*)d_out;                    // (B,D,H,W) fp32 = CG solution

  // Workspace: [0,32KB) double accumulators; then r, P0, P1, Ap (21 MB each).
  // Whole working set (~105 MB incl. x) stays resident in the 192 MB L2.
  double* acc = (double*)d_ws;
  float* base = (float*)((char*)d_ws + 32768);
  float* r  = base;
  float* P[2] = { base + NTOT, base + 2 * NTOT };
  float* Ap = base + 3 * NTOT;

  dim3 tg(W_ / TW, H_ / TH, B_);               // 5 x 8 x 16 = 640 blocks

  k_zero<<<1, 256, 0, stream>>>(acc);
  k_init<<<tg, 256, 0, stream>>>(ae, wxwy, x, r, P[0], acc);

  for (int k = 0; k < MAXITER; ++k) {
    const float* p_in  = P[k & 1];
    float*       p_out = P[(k + 1) & 1];
    int bn = (k == 0) ? ACC_ZERO : ACC_RR(k);       // beta numerator
    int bd = (k == 0) ? ACC_RR(0) : ACC_RR(k - 1);  // beta denominator
    int an = (k == 0) ? ACC_ZERO : ACC_RR(k - 1);   // alpha_{k-1} numerator
    int ad = (k == 0) ? ACC_RR(0) : ACC_PAP(k - 1); // alpha_{k-1} denominator
    k_matvec<<<tg, 256, 0, stream>>>(r, p_in, p_out, Ap, x, wxwy, acc,
                                     bn, bd, an, ad, ACC_PAP(k), (k > 0));
    k_update<<<UPD_BLOCKS, 256, 0, stream>>>(
        r, Ap, acc, ACC_RR(k), ACC_PAP(k), ACC_RR(k + 1));
  }
  // last lagged term: x += alpha_{499} * p_{499}  (p_out of iter 499 = P[0])
  k_xfinal<<<UPD_BLOCKS, 256, 0, stream>>>(
      x, P[MAXITER & 1], acc, ACC_RR(MAXITER - 1), ACC_PAP(MAXITER - 1));
}